// JacobiPool_8263517077811
// MI455X (gfx1250) — compile-verified
//
#include <hip/hip_runtime.h>
#include <hip/hip_bf16.h>
#include <math.h>

typedef __attribute__((ext_vector_type(2))) float v2f;
typedef __attribute__((ext_vector_type(8))) float v8f;

#define IN_CH 128
#define NEG_SLOPE 0.2f

// ---------- helpers: order-preserving float<->uint key for atomic max ----------
__device__ __forceinline__ unsigned fkey(float f) {
    unsigned u = __float_as_uint(f);
    return u ^ ((u >> 31) ? 0xFFFFFFFFu : 0x80000000u);
}
__device__ __forceinline__ float funkey(unsigned k) {
    unsigned b = (k & 0x80000000u) ? (k ^ 0x80000000u) : ~k;
    return __uint_as_float(b);
}

// ---------- kernel 0: init per-node accumulators ----------
__global__ void gat_init(unsigned* __restrict__ mkey, float* __restrict__ ssum,
                         float* __restrict__ score, const float* __restrict__ bias, int N) {
    int i = blockIdx.x * blockDim.x + threadIdx.x;
    if (i < N) {
        mkey[i] = fkey(-INFINITY);
        ssum[i] = 0.0f;
        score[i] = bias[0];
    }
}

// ---------- kernel 1: h = x @ W via V_WMMA_F32_16X16X4_F32, one wave = 16 nodes ----------
// A layout (32-bit 16x4, ISA 7.12.2): lane L<16 holds row M=L, K={0,1}; lane L>=16 holds
// row M=L-16, K={2,3}. B is W[k..k+3] broadcast across all 16 columns, so every column of
// D equals h for the 16-node tile. Accumulate K=128 in 32 steps.
__global__ void gat_gemv_wmma(const float* __restrict__ x, const float* __restrict__ W,
                              const float* __restrict__ att_src, const float* __restrict__ att_dst,
                              float* __restrict__ h, float* __restrict__ as_out,
                              float* __restrict__ ad_out, int N) {
    int lane = threadIdx.x & 31;
    int wave = blockIdx.x * (blockDim.x >> 5) + (threadIdx.x >> 5);
    int base = wave * 16;
    if (base >= N) return;              // wave-uniform: EXEC stays all-ones for WMMA

    int m    = lane & 15;
    int koff = (lane >> 4) << 1;        // 0 for lanes 0-15, 2 for lanes 16-31
    const float* xrow = x + (size_t)(base + m) * IN_CH + koff;
    const float* wp   = W + koff;

    v8f c = {};
#pragma unroll
    for (int k = 0; k < IN_CH; k += 4) {
        v2f a = *(const v2f*)(xrow + k);   // x[M][k+koff], x[M][k+koff+1]
        v2f b = *(const v2f*)(wp + k);     // W[k+koff],    W[k+koff+1]  (broadcast over N)
        c = __builtin_amdgcn_wmma_f32_16x16x4_f32(
            /*neg_a=*/false, a, /*neg_b=*/false, b,
            /*c_mod=*/(short)0, c, /*reuse_a=*/false, /*reuse_b=*/false);
    }

    // D column 0: lane 0 has M=0..7 in c[0..7]; lane 16 has M=8..15 in c[0..7]
    if (m == 0) {
        int off = (lane >> 4) * 8;
        float as = att_src[0], ad = att_dst[0];
#pragma unroll
        for (int r = 0; r < 8; ++r) {
            float hv = c[r];
            int idx = base + off + r;
            h[idx] = hv;
            as_out[idx] = hv * as;
            ad_out[idx] = hv * ad;
        }
    }
}

// scalar tail (only launched if N % 16 != 0; for N=100000 it is not)
__global__ void gat_gemv_tail(const float* __restrict__ x, const float* __restrict__ W,
                              const float* __restrict__ att_src, const float* __restrict__ att_dst,
                              float* __restrict__ h, float* __restrict__ as_out,
                              float* __restrict__ ad_out, int start, int N) {
    int i = start + blockIdx.x * blockDim.x + threadIdx.x;
    if (i >= N) return;
    const float* xr = x + (size_t)i * IN_CH;
    float acc = 0.0f;
    for (int k = 0; k < IN_CH; ++k) acc += xr[k] * W[k];
    h[i] = acc; as_out[i] = acc * att_src[0]; ad_out[i] = acc * att_dst[0];
}

// ---------- kernel 2: logits + segment max + edge copy ----------
__global__ void gat_pass1(const long long* __restrict__ ei,
                          const float* __restrict__ a_src, const float* __restrict__ a_dst,
                          float* __restrict__ ebuf,          // alpha slot of d_out (scratch)
                          float* __restrict__ edge0, float* __restrict__ edge1,
                          unsigned* __restrict__ mkey, int E, int M) {
    int i = blockIdx.x * blockDim.x + threadIdx.x;
    if (i >= M) return;
    int src, dst;
    if (i < E) { src = (int)ei[i]; dst = (int)ei[(size_t)E + i]; }
    else       { src = dst = i - E; }                 // self-loops appended
    float t = a_src[src] + a_dst[dst];
    float e = (t > 0.0f) ? t : NEG_SLOPE * t;
    ebuf[i]  = e;
    edge0[i] = (float)src;                            // exact: indices < 2^24
    edge1[i] = (float)dst;
    atomicMax(&mkey[dst], fkey(e));
}

// ---------- kernel 3: exp(e - m) + segment sum ----------
__global__ void gat_pass2(const float* __restrict__ edge1, const unsigned* __restrict__ mkey,
                          float* __restrict__ ebuf, float* __restrict__ ssum, int M) {
    int i = blockIdx.x * blockDim.x + threadIdx.x;
    if (i >= M) return;
    int dst = (int)edge1[i];
    float ex = __expf(ebuf[i] - funkey(mkey[dst]));
    ebuf[i] = ex;
    atomicAdd(&ssum[dst], ex);
}

// ---------- kernel 4: alpha + weighted aggregation ----------
__global__ void gat_pass3(const float* __restrict__ edge0, const float* __restrict__ edge1,
                          const float* __restrict__ ssum, const float* __restrict__ h,
                          float* __restrict__ ebuf,       // becomes final alpha
                          float* __restrict__ score, int M) {
    int i = blockIdx.x * blockDim.x + threadIdx.x;
    if (i >= M) return;
    int src = (int)edge0[i];
    int dst = (int)edge1[i];
    float alpha = ebuf[i] / (ssum[dst] + 1e-16f);
    ebuf[i] = alpha;
    atomicAdd(&score[dst], h[src] * alpha);
}

extern "C" void kernel_launch(void* const* d_in, const int* in_sizes, int n_in,
                              void* d_out, int out_size, void* d_ws, size_t ws_size,
                              hipStream_t stream) {
    const float*     x       = (const float*)d_in[0];
    const long long* ei      = (const long long*)d_in[1];
    const float*     W       = (const float*)d_in[2];
    const float*     att_src = (const float*)d_in[3];
    const float*     att_dst = (const float*)d_in[4];
    const float*     bias    = (const float*)d_in[5];

    const int N = in_sizes[0] / IN_CH;   // 100000
    const int E = in_sizes[1] / 2;       // 3200000
    const int M = E + N;                 // messages incl. self-loops

    // d_out layout (floats): score[N] | edge0[M] | edge1[M] | alpha[M]
    float* score = (float*)d_out;
    float* edge0 = score + N;
    float* edge1 = edge0 + M;
    float* ebuf  = edge1 + M;            // e -> exp(e-m) -> alpha (in place)

    // workspace layout: h[N] | a_src[N] | a_dst[N] | mkey[N] | ssum[N]  (~2 MB)
    float*    h    = (float*)d_ws;
    float*    a_s  = h + N;
    float*    a_d  = a_s + N;
    unsigned* mkey = (unsigned*)(a_d + N);
    float*    ssum = (float*)(mkey + N);

    const int B = 256;

    gat_init<<<(N + B - 1) / B, B, 0, stream>>>(mkey, ssum, score, bias, N);

    int ntile = N / 16;                              // full 16-node wave tiles
    int waves_per_block = B / 32;                    // 8
    int gblocks = (ntile + waves_per_block - 1) / waves_per_block;
    if (gblocks > 0)
        gat_gemv_wmma<<<gblocks, B, 0, stream>>>(x, W, att_src, att_dst, h, a_s, a_d, ntile * 16);
    int rem = N - ntile * 16;
    if (rem > 0)
        gat_gemv_tail<<<(rem + B - 1) / B, B, 0, stream>>>(x, W, att_src, att_dst, h, a_s, a_d,
                                                           ntile * 16, N);

    int mb = (M + B - 1) / B;
    gat_pass1<<<mb, B, 0, stream>>>(ei, a_s, a_d, ebuf, edge0, edge1, mkey, E, M);
    gat_pass2<<<mb, B, 0, stream>>>(edge1, mkey, ebuf, ssum, M);
    gat_pass3<<<mb, B, 0, stream>>>(edge0, edge1, ssum, h, ebuf, score, M);
}